// singular_spectrum_operator_14946486190629
// MI455X (gfx1250) — compile-verified
//
#include <hip/hip_runtime.h>
#include <cstdint>

typedef uint32_t u32;
typedef uint64_t u64;
typedef __attribute__((ext_vector_type(4))) u32   v4u;
typedef __attribute__((ext_vector_type(8))) int   v8i;
typedef __attribute__((ext_vector_type(4))) int   v4i;
typedef __attribute__((ext_vector_type(2))) float v2f;
typedef __attribute__((ext_vector_type(8))) float v8f;

#define BATCH 256
#define CH    64
#define NLEN  2048
#define TILE  256                   // columns (n) per TDM tile
#define NT    (NLEN / TILE)         // 8 tiles
#define TILE_ELEMS (CH * TILE)      // 16384 floats = 64 KB
#define TILE_BYTES (TILE_ELEMS * 4)

// Issue a TDM 2D tile load: [rows x cols] f32, row stride = stride_elems,
// global tile start = gptr, LDS destination byte offset = lds_byte.
__device__ __forceinline__ void tdm_load_tile_2d(u32 lds_byte, const void* gptr,
                                                 u32 rows, u32 cols,
                                                 u32 stride_elems) {
  u64 ga = (u64)(uintptr_t)gptr;
  v4u g0;
  g0[0] = 1u;                                   // count=1, is_restore=0, gather off
  g0[1] = lds_byte;                             // lds_addr (bytes)
  g0[2] = (u32)(ga & 0xFFFFFFFFull);            // global_addr[31:0]
  g0[3] = (u32)((ga >> 32) & 0x01FFFFFFull)     // global_addr[56:32]
        | 0x80000000u;                          // type = 2 ("image") in bits 127:126
  v8i g1;
  g1[0] = (int)0x00020000u;                     // workgroup_mask=0, data_size=2 (4B)
  g1[1] = (int)((cols & 0xFFFFu) << 16);        // tensor_dim0[15:0]   @ bits 63:48
  g1[2] = (int)(((cols >> 16) & 0xFFFFu)        // tensor_dim0[31:16]  @ bits 79:64
        | ((rows & 0xFFFFu) << 16));            // tensor_dim1[15:0]   @ bits 95:80
  g1[3] = (int)(((rows >> 16) & 0xFFFFu)        // tensor_dim1[31:16]  @ bits 111:96
        | ((cols & 0xFFFFu) << 16));            // tile_dim0           @ bits 127:112
  g1[4] = (int)(rows & 0xFFFFu);                // tile_dim1 (tile_dim2 = 0)
  g1[5] = (int)stride_elems;                    // tensor_dim0_stride[31:0]
  g1[6] = 0;                                    // stride hi / dim1_stride lo (unused 2D)
  g1[7] = 0;
  v4i z4; z4[0] = z4[1] = z4[2] = z4[3] = 0;    // groups 2/3 unused (2D tensor)
  v8i z8;
  #pragma unroll
  for (int i = 0; i < 8; ++i) z8[i] = 0;
  __builtin_amdgcn_tensor_load_to_lds(g0, g1, z4, z4, z8, 0);
}

__global__ void __launch_bounds__(256)
ssa_softmax_gate(const float* __restrict__ x, float* __restrict__ out) {
  extern __shared__ float smem[];
  float* gate = smem + 2 * TILE_ELEMS;          // 2048 floats
  float* redM = gate + NLEN;                    // 8 wave maxima
  float* redS = redM + 8;                       // 8 wave partial sums
  float* redT = redS + 8;                       // 1 total

  const int tid  = threadIdx.x;
  const int lane = tid & 31;
  const int wid  = tid >> 5;
  const int b    = blockIdx.x;
  const size_t base = (size_t)b * CH * NLEN;
  const float* xb = x + base;

  const u32 lds_base = (u32)(uintptr_t)(void*)smem;  // low 32 bits = LDS offset

  // ---------- Phase 1: xm[n] = max_c x[b,c,n] via double-buffered TDM ----------
  if (tid < 32) {
    tdm_load_tile_2d(lds_base, xb, CH, TILE, NLEN);
  }
  for (int t = 0; t < NT; ++t) {
    if (tid < 32) {
      if (t + 1 < NT) {
        tdm_load_tile_2d(lds_base + (u32)((t + 1) & 1) * TILE_BYTES,
                         xb + (size_t)(t + 1) * TILE, CH, TILE, NLEN);
        __builtin_amdgcn_s_wait_tensorcnt(1);   // tile t landed; t+1 in flight
      } else {
        __builtin_amdgcn_s_wait_tensorcnt(0);
      }
    }
    __syncthreads();                            // tile t visible to all waves
    const float* tb = smem + (t & 1) * TILE_ELEMS;
    float m = tb[tid];
    #pragma unroll
    for (int r = 1; r < CH; ++r) m = fmaxf(m, tb[r * TILE + tid]);
    gate[t * TILE + tid] = m;
    __syncthreads();                            // buffer free before reuse at t+2
  }

  // ---------- Phase 2: softmax over gate[0..NLEN) ----------
  float v[8];
  #pragma unroll
  for (int k = 0; k < 8; ++k) v[k] = gate[k * 256 + tid];
  float m = v[0];
  #pragma unroll
  for (int k = 1; k < 8; ++k) m = fmaxf(m, v[k]);
  #pragma unroll
  for (int off = 16; off > 0; off >>= 1) m = fmaxf(m, __shfl_xor(m, off));
  if (lane == 0) redM[wid] = m;
  __syncthreads();
  float mx = redM[0];
  #pragma unroll
  for (int w = 1; w < 8; ++w) mx = fmaxf(mx, redM[w]);

  float e[8];
  float p = 0.0f;
  #pragma unroll
  for (int k = 0; k < 8; ++k) { e[k] = __expf(v[k] - mx); p += e[k]; }
  #pragma unroll
  for (int off = 16; off > 0; off >>= 1) p += __shfl_xor(p, off);
  if (lane == 0) redS[wid] = p;
  __syncthreads();

  // Final 8-value sum reduction on wave 0 via WMMA dot-with-ones:
  // A = ones(16x4), B[0][j] = s_j (j<8) else 0  =>  D[0][j] = s_j; shfl-sum 16 cols.
  if (tid < 32) {                               // EXEC all-ones for this wave
    float s = (lane < 8) ? redS[lane] : 0.0f;
    v2f aa = {1.0f, 1.0f};
    v2f bb = {s, 0.0f};
    v8f cc = {};
    v8f dd = __builtin_amdgcn_wmma_f32_16x16x4_f32(
        false, aa, false, bb, (short)0, cc, false, false);
    float tot = dd[0];
    #pragma unroll
    for (int off = 8; off > 0; off >>= 1) tot += __shfl_xor(tot, off);
    if (lane == 0) redT[0] = tot;
  }
  __syncthreads();
  const float inv = 1.0f / redT[0];
  #pragma unroll
  for (int k = 0; k < 8; ++k) gate[k * 256 + tid] = e[k] * inv;
  __syncthreads();

  // ---------- Phase 3: out[b,c,n] = gate[n] * x[b,c,n] (streaming, L2 reread) ----------
  const float4* gv = (const float4*)gate;
  const float4 ga4 = gv[tid];
  const float4 gb4 = gv[tid + 256];
  #pragma unroll 4
  for (int c = 0; c < CH; ++c) {
    const float4* s4 = (const float4*)(xb + (size_t)c * NLEN);
    float4*       d4 = (float4*)(out + base + (size_t)c * NLEN);
    float4 a0 = s4[tid];
    float4 a1 = s4[tid + 256];
    d4[tid]       = make_float4(a0.x * ga4.x, a0.y * ga4.y, a0.z * ga4.z, a0.w * ga4.w);
    d4[tid + 256] = make_float4(a1.x * gb4.x, a1.y * gb4.y, a1.z * gb4.z, a1.w * gb4.w);
  }
}

extern "C" void kernel_launch(void* const* d_in, const int* in_sizes, int n_in,
                              void* d_out, int out_size, void* d_ws, size_t ws_size,
                              hipStream_t stream) {
  const float* x = (const float*)d_in[0];
  float* out = (float*)d_out;
  const size_t smem_bytes = (size_t)(2 * TILE_ELEMS + NLEN + 32) * sizeof(float);
  (void)hipFuncSetAttribute((const void*)ssa_softmax_gate,
                            hipFuncAttributeMaxDynamicSharedMemorySize,
                            (int)smem_bytes);
  ssa_softmax_gate<<<dim3(BATCH), dim3(256), smem_bytes, stream>>>(x, out);
}